// MultiScaleWindowGenerator_61091614819052
// MI455X (gfx1250) — compile-verified
//
#include <hip/hip_runtime.h>
#include <hip/hip_bf16.h>
#include <stdint.h>

// MultiScaleWindowGenerator: out_L[b,n,c,l] = in[b,c, n - L/2 + l] (0 outside [0,N))
// B=2, C=64, N=512, L in {256, 1024, 512, 512}; output ~604MB fp32 -> store-BW bound.

#define BLOCK   256
#define NT      64     // n values per block
#define N_SEQ   512
#define C_DIM   64
#define B_DIM   2
#define LDS_PAD 512
#define LDS_FLOATS (LDS_PAD + N_SEQ + LDS_PAD)   // 1536 floats = 6 KB

typedef float vfloat4 __attribute__((ext_vector_type(4)));
typedef int   vint4   __attribute__((ext_vector_type(4)));

// Address-space-qualified pointer types for the async-copy builtin.
typedef __attribute__((address_space(1))) vint4* as1_vint4_ptr;
typedef __attribute__((address_space(3))) vint4* as3_vint4_ptr;

#if defined(__has_builtin)
#  if __has_builtin(__builtin_amdgcn_global_load_async_to_lds_b128)
#    define HAVE_ASYNC_B128 1
#  endif
#  if __has_builtin(__builtin_amdgcn_s_wait_asynccnt)
#    define HAVE_WAIT_ASYNC 1
#  endif
#endif

__global__ __launch_bounds__(BLOCK)
void msw_kernel(const float* __restrict__ in, float* __restrict__ out,
                int half, int l4shift) {
    __shared__ float lds[LDS_FLOATS];
    const int tid = threadIdx.x;
    const int bc  = blockIdx.x;          // b*C + c, 0..127
    const int b   = bc >> 6;
    const int c   = bc & 63;
    const int n0  = blockIdx.y * NT;

    // Zero the left/right pads (512 floats each); middle is overwritten below.
    lds[tid]                           = 0.0f;
    lds[tid + BLOCK]                   = 0.0f;
    lds[LDS_PAD + N_SEQ + tid]         = 0.0f;
    lds[LDS_PAD + N_SEQ + BLOCK + tid] = 0.0f;

    const float* row = in + (size_t)(b * C_DIM + c) * N_SEQ;
    float* dst = lds + LDS_PAD;

#if HAVE_ASYNC_B128
    // CDNA5 async copy: 128 lanes x 16B = full 2KB row, global -> LDS (ASYNCcnt).
    if (tid < (N_SEQ / 4)) {
        as1_vint4_ptr g = (as1_vint4_ptr)(uintptr_t)(row + tid * 4);
        as3_vint4_ptr l = (as3_vint4_ptr)(uint32_t)(uintptr_t)(dst + tid * 4);
        __builtin_amdgcn_global_load_async_to_lds_b128(g, l, 0, 0);
    }
#  if HAVE_WAIT_ASYNC
    __builtin_amdgcn_s_wait_asynccnt(0);
#  else
    asm volatile("s_wait_asynccnt 0" ::: "memory");
#  endif
#else
    // Fallback: plain cooperative copy (256 threads x float2 = 2KB).
    {
        float2 v = ((const float2*)row)[tid];
        ((float2*)dst)[tid] = v;
    }
#endif
    __syncthreads();

    // Each iteration writes one 16B quad of an output row.
    // out flat index = ((b*N + n)*C + c)*L + l ; L = 4 << l4shift.
    const int   qmask = (1 << l4shift) - 1;
    const int   total = NT << l4shift;            // NT * L/4 quads per block
    const float* lp   = lds + LDS_PAD;

    for (int idx = tid; idx < total; idx += BLOCK) {
        const int n   = n0 + (idx >> l4shift);
        const int l4  = (idx & qmask) << 2;
        const int src = n - half + l4;            // in [-512, 1022], pad-covered
        vfloat4 v;
        v.x = lp[src + 0];
        v.y = lp[src + 1];
        v.z = lp[src + 2];
        v.w = lp[src + 3];
        const size_t off =
            ((size_t)((b * N_SEQ + n) * C_DIM + c) << (l4shift + 2)) + (size_t)l4;
        __builtin_nontemporal_store(v, (vfloat4*)(out + off));
    }
}

extern "C" void kernel_launch(void* const* d_in, const int* in_sizes, int n_in,
                              void* d_out, int out_size, void* d_ws, size_t ws_size,
                              hipStream_t stream) {
    (void)in_sizes; (void)n_in; (void)out_size; (void)d_ws; (void)ws_size;
    const float* in  = (const float*)d_in[0];
    float*       out = (float*)d_out;

    const int Ls[4] = {256, 1024, 512, 512};      // floor(N*sqrt(s)), floor(N/sqrt(s)) for s in {0.25, 1.0}
    dim3 grid(B_DIM * C_DIM, N_SEQ / NT);
    dim3 block(BLOCK);

    size_t off = 0;
    for (int i = 0; i < 4; ++i) {
        const int L      = Ls[i];
        const int half   = L / 2;
        const int l4shft = __builtin_ctz(L / 4);  // L/4 is a power of two: 64/256/128
        msw_kernel<<<grid, block, 0, stream>>>(in, out + off, half, l4shft);
        off += (size_t)B_DIM * N_SEQ * C_DIM * L;
    }
}